// RepulsionLoss_30064771072407
// MI455X (gfx1250) — compile-verified
//
#include <hip/hip_runtime.h>
#include <math.h>

typedef __attribute__((ext_vector_type(2))) float v2f;
typedef __attribute__((ext_vector_type(8))) float v8f;

#define NB     16        // batches
#define NPTS   4096      // points per batch
#define KNN    5
#define RADIUS_C 0.07f
#define H_C      0.03f
#define EPS_C    1e-12f
#define BLOCKS_PER_BATCH 32   // 32 blocks * 8 waves * 16 queries = 4096

__global__ void zero_out_kernel(float* out, int n) {
    int i = blockIdx.x * blockDim.x + threadIdx.x;
    if (i < n) out[i] = 0.0f;
}

__global__ void __launch_bounds__(256)
repulsion_loss_kernel(const float* __restrict__ pred, float* __restrict__ out)
{
    extern __shared__ float smem[];
    float* s_x    = smem;              // [4096]
    float* s_y    = smem + NPTS;       // [4096]
    float* s_z    = smem + 2 * NPTS;   // [4096]
    float* s_part = smem + 3 * NPTS;   // [8]

    const int b    = blockIdx.x / BLOCKS_PER_BATCH;
    const int nblk = blockIdx.x % BLOCKS_PER_BATCH;
    const int tid  = threadIdx.x;
    const int lane = tid & 31;
    const int wave = tid >> 5;

    // ---- stage this batch's points into LDS (SoA) ----
    const float* pb = pred + (size_t)b * NPTS * 3;
    for (int i = tid; i < NPTS; i += 256) {
        float x = pb[i * 3 + 0];
        float y = pb[i * 3 + 1];
        float z = pb[i * 3 + 2];
        s_x[i] = x; s_y[i] = y; s_z[i] = z;
    }
    __syncthreads();

    // ---- B fragment: 16 query points per wave (constant over m loop) ----
    // B[k,n]: k0=-2x_n, k1=-2y_n, k2=-2z_n, k3=1   (lanes<16: k0,k1; lanes>=16: k2,k3)
    const bool hi = (lane >= 16);
    const int  nl = lane & 15;
    const int  n0 = nblk * 128 + wave * 16;
    const int  nq = n0 + nl;
    const float xq = s_x[nq], yq = s_y[nq], zq = s_z[nq];
    const float xxq = xq * xq + yq * yq + zq * zq;

    v2f bf;
    bf[0] = hi ? (-2.0f * zq) : (-2.0f * xq);
    bf[1] = hi ? 1.0f         : (-2.0f * yq);

    // C fragment: C[m,n] = |q_n|^2 -> per-lane broadcast of own query norm
    v8f cfrag;
    cfrag[0] = xxq; cfrag[1] = xxq; cfrag[2] = xxq; cfrag[3] = xxq;
    cfrag[4] = xxq; cfrag[5] = xxq; cfrag[6] = xxq; cfrag[7] = xxq;

    // running 6 smallest squared distances (ascending), per lane stream
    float t0 = 3.0e38f, t1 = 3.0e38f, t2 = 3.0e38f,
          t3 = 3.0e38f, t4 = 3.0e38f, t5 = 3.0e38f;

#define INS6(VAL) do { float vv = (VAL), mn_;                         \
        mn_ = fminf(t0, vv); vv = fmaxf(t0, vv); t0 = mn_;            \
        mn_ = fminf(t1, vv); vv = fmaxf(t1, vv); t1 = mn_;            \
        mn_ = fminf(t2, vv); vv = fmaxf(t2, vv); t2 = mn_;            \
        mn_ = fminf(t3, vv); vv = fmaxf(t3, vv); t3 = mn_;            \
        mn_ = fminf(t4, vv); vv = fmaxf(t4, vv); t4 = mn_;            \
        t5 = fminf(t5, vv); } while (0)

    // ---- sweep all candidate tiles: one WMMA per 16x16 dist^2 block ----
#pragma unroll 4
    for (int m0 = 0; m0 < NPTS; m0 += 16) {
        const int ml = m0 + nl;
        const float xm = s_x[ml], ym = s_y[ml], zm = s_z[ml];
        const float xxm = fmaf(xm, xm, fmaf(ym, ym, zm * zm));

        // A[m,k]: k0=x_m, k1=y_m, k2=z_m, k3=|p_m|^2
        v2f af;
        af[0] = hi ? zm  : xm;
        af[1] = hi ? xxm : ym;

        // D[m,n] = |p_m|^2 - 2 p_m.q_n + |q_n|^2  (full squared distance)
        v8f d = __builtin_amdgcn_wmma_f32_16x16x4_f32(
            false, af, false, bf, (short)0, cfrag, false, false);

        INS6(d[0]); INS6(d[1]); INS6(d[2]); INS6(d[3]);
        INS6(d[4]); INS6(d[5]); INS6(d[6]); INS6(d[7]);
    }

    // ---- merge lane-pair lists (rows 0-7 stream vs rows 8-15 stream) ----
    // snapshot partner values BEFORE inserting (both lanes mutate in lockstep)
    const float o0 = __shfl_xor(t0, 16, 32);
    const float o1 = __shfl_xor(t1, 16, 32);
    const float o2 = __shfl_xor(t2, 16, 32);
    const float o3 = __shfl_xor(t3, 16, 32);
    const float o4 = __shfl_xor(t4, 16, 32);
    const float o5 = __shfl_xor(t5, 16, 32);
    INS6(o0); INS6(o1); INS6(o2); INS6(o3); INS6(o4); INS6(o5);

    // ---- loss tail: drop self (t0 = global min), use next 5 ----
    const float h2inv = 1.0f / (H_C * H_C);
    float acc = 0.0f;
    if (!hi) {
#define CONTRIB(D2) do { float dc = fmaxf((D2), EPS_C);               \
        float dist = sqrtf(dc);                                       \
        float w = __expf(-dc * h2inv);                                \
        acc += (RADIUS_C - dist) * w; } while (0)
        CONTRIB(t1); CONTRIB(t2); CONTRIB(t3); CONTRIB(t4); CONTRIB(t5);
    }

    // ---- reduce: wave -> LDS -> one atomic per block ----
    acc += __shfl_xor(acc, 1, 32);
    acc += __shfl_xor(acc, 2, 32);
    acc += __shfl_xor(acc, 4, 32);
    acc += __shfl_xor(acc, 8, 32);
    acc += __shfl_xor(acc, 16, 32);
    if (lane == 0) s_part[wave] = acc;
    __syncthreads();
    if (tid == 0) {
        float s = 0.0f;
#pragma unroll
        for (int i = 0; i < 8; ++i) s += s_part[i];
        atomicAdd(&out[b], s * (1.0f / ((float)NPTS * (float)KNN)));
    }
}

extern "C" void kernel_launch(void* const* d_in, const int* in_sizes, int n_in,
                              void* d_out, int out_size, void* d_ws, size_t ws_size,
                              hipStream_t stream) {
    (void)in_sizes; (void)n_in; (void)d_ws; (void)ws_size;
    const float* pred = (const float*)d_in[0];
    float* out = (float*)d_out;

    zero_out_kernel<<<1, 32, 0, stream>>>(out, out_size);

    const size_t smem_bytes = (3 * NPTS + 8) * sizeof(float);   // 48 KB + partials
    repulsion_loss_kernel<<<dim3(NB * BLOCKS_PER_BATCH), dim3(256),
                            smem_bytes, stream>>>(pred, out);
}